// CausalConv1dUpdateModel_8014408974841
// MI455X (gfx1250) — compile-verified
//
#include <hip/hip_runtime.h>
#include <stdint.h>

// Problem constants (B, S, D, W, NUM_SLOTS) = (8192, 1, 4096, 4, 16384), S==1.
namespace {
constexpr int kB      = 8192;
constexpr int kD      = 4096;
constexpr int kSlots  = 16384;
constexpr int kWm1    = 3;           // W - 1 state rows
constexpr int kThreads = 256;
constexpr int kVec    = 4;           // float4 per thread
constexpr int kChunk  = kThreads * kVec;   // 1024 channels per block
}

typedef float v4f __attribute__((ext_vector_type(4)));
typedef __attribute__((address_space(3))) float lds_f;

// ---------------- inverse-index build (slot -> batch, or -1) ----------------
__global__ void k_init_inverse(int* __restrict__ inv) {
  int t = blockIdx.x * blockDim.x + threadIdx.x;
  if (t < kSlots) inv[t] = -1;
}

__global__ void k_scatter_inverse(const int* __restrict__ idx, int* __restrict__ inv) {
  int b = blockIdx.x * blockDim.x + threadIdx.x;
  if (b < kB) inv[idx[b]] = b;
}

// ---------------- fused conv-update + state shift + state copy --------------
// grid = (kD / kChunk, kSlots); each block handles 1024 channels of one slot.
__global__ __launch_bounds__(kThreads) void k_conv_update(
    const float* __restrict__ x,       // [B, D]
    const float* __restrict__ cs,      // [NUM_SLOTS, 3, D] (input, not mutated)
    const float* __restrict__ w,       // [4, D]
    const float* __restrict__ bias,    // [D]
    const int*   __restrict__ inv,     // [NUM_SLOTS] slot -> batch or -1
    float* __restrict__ out,           // [B, D]
    float* __restrict__ cs_out)        // [NUM_SLOTS, 3, D]
{
  __shared__ float lbuf[kWm1][kChunk];            // 12 KB, thread-private slices
  const int    slot = blockIdx.y;
  const int    tid  = threadIdx.x;
  const int    d    = blockIdx.x * kChunk + tid * kVec;
  const int    b    = inv[slot];                  // uniform within the block
  const size_t srow = (size_t)slot * (kWm1 * kD) + d;

  if (b >= 0) {
    // ---- compute path: out = silu(bias + c0*w0 + c1*w1 + c2*w2 + x*w3) ----
    // conv_state and x are streamed once -> non-temporal loads.
    const v4f c0 = __builtin_nontemporal_load((const v4f*)(cs + srow));
    const v4f c1 = __builtin_nontemporal_load((const v4f*)(cs + srow + kD));
    const v4f c2 = __builtin_nontemporal_load((const v4f*)(cs + srow + 2 * kD));
    const v4f xv = __builtin_nontemporal_load((const v4f*)(x + (size_t)b * kD + d));
    // weight/bias are reused by every slot -> regular (cacheable) loads.
    const v4f w0 = *(const v4f*)(w + 0 * kD + d);
    const v4f w1 = *(const v4f*)(w + 1 * kD + d);
    const v4f w2 = *(const v4f*)(w + 2 * kD + d);
    const v4f w3 = *(const v4f*)(w + 3 * kD + d);
    const v4f bb = *(const v4f*)(bias + d);

    v4f o = bb + c0 * w0 + c1 * w1 + c2 * w2 + xv * w3;
#pragma unroll
    for (int i = 0; i < kVec; ++i) {
      const float oi = o[i];
      o[i] = oi / (1.0f + __expf(-oi));           // silu = o * sigmoid(o)
    }
    // Outputs are never re-read -> non-temporal B128 stores.
    __builtin_nontemporal_store(o,  (v4f*)(out + (size_t)b * kD + d));
    __builtin_nontemporal_store(c1, (v4f*)(cs_out + srow));           // shift
    __builtin_nontemporal_store(c2, (v4f*)(cs_out + srow + kD));
    __builtin_nontemporal_store(xv, (v4f*)(cs_out + srow + 2 * kD));
  } else {
    // ---- untouched slot: bulk copy via gfx1250 async global<->LDS DMA ----
    // Each lane moves 16B per row through its private LDS slice; no VGPR
    // round-trip, tracked on ASYNCcnt.
    const uint64_t g_src = (uint64_t)(uintptr_t)(cs + srow);
    const uint64_t g_dst = (uint64_t)(uintptr_t)(cs_out + srow);
    const uint32_t l0 = (uint32_t)(uintptr_t)(lds_f*)&lbuf[0][tid * kVec];
    const uint32_t l1 = l0 + (uint32_t)(kChunk * sizeof(float));
    const uint32_t l2 = l1 + (uint32_t)(kChunk * sizeof(float));
    const uint64_t rowB = (uint64_t)kD * sizeof(float);

    asm volatile("global_load_async_to_lds_b128 %0, %1, off"
                 :: "v"(l0), "v"(g_src) : "memory");
    asm volatile("global_load_async_to_lds_b128 %0, %1, off"
                 :: "v"(l1), "v"(g_src + rowB) : "memory");
    asm volatile("global_load_async_to_lds_b128 %0, %1, off"
                 :: "v"(l2), "v"(g_src + 2 * rowB) : "memory");
    asm volatile("s_wait_asynccnt 0x0" ::: "memory");
    asm volatile("global_store_async_from_lds_b128 %0, %1, off"
                 :: "v"(g_dst), "v"(l0) : "memory");
    asm volatile("global_store_async_from_lds_b128 %0, %1, off"
                 :: "v"(g_dst + rowB), "v"(l1) : "memory");
    asm volatile("global_store_async_from_lds_b128 %0, %1, off"
                 :: "v"(g_dst + 2 * rowB), "v"(l2) : "memory");
    // S_ENDPGM performs an implicit wait-idle, so outstanding async stores
    // are drained before the wave retires.
  }
}

extern "C" void kernel_launch(void* const* d_in, const int* in_sizes, int n_in,
                              void* d_out, int out_size, void* d_ws, size_t ws_size,
                              hipStream_t stream) {
  (void)in_sizes; (void)n_in; (void)out_size; (void)ws_size;
  const float* x    = (const float*)d_in[0];   // (8192, 1, 4096) f32
  const float* cs   = (const float*)d_in[1];   // (16384, 3, 4096) f32
  const float* w    = (const float*)d_in[2];   // (4, 4096) f32
  const float* bias = (const float*)d_in[3];   // (4096,) f32
  const int*   idx  = (const int*)d_in[4];     // (8192,) i32

  float* out    = (float*)d_out;               // first output: (8192, 1, 4096)
  float* cs_out = out + (size_t)kB * kD;       // second output: (16384, 3, 4096)
  int*   inv    = (int*)d_ws;                  // 64 KB inverse map

  k_init_inverse<<<kSlots / kThreads, kThreads, 0, stream>>>(inv);
  k_scatter_inverse<<<kB / kThreads, kThreads, 0, stream>>>(idx, inv);

  dim3 grid(kD / kChunk, kSlots);              // (4, 16384)
  k_conv_update<<<grid, kThreads, 0, stream>>>(x, cs, w, bias, inv, out, cs_out);
}